// MultiheadAttention_80058190397962
// MI455X (gfx1250) — compile-verified
//
#include <hip/hip_runtime.h>
#include <hip/hip_bf16.h>

typedef __attribute__((ext_vector_type(2))) float v2f;
typedef __attribute__((ext_vector_type(8))) float v8f;

#define N_NODES 50000
#define E_EDGES 1600000
#define NHEAD   4
#define DHEAD   16
#define F_IN    128
#define HD      64   // NHEAD * DHEAD

// ---------------------------------------------------------------------------
// Kernel 1: Q/K/V projections via V_WMMA_F32_16X16X4_F32.
// One wave computes a 16x64 output tile for one of {Q,K,V}.
// A: 16x4 fp32 fragment (2 VGPRs), B: 4x16 fp32 fragment (2 VGPRs), C/D: 8 VGPRs.
// Lane layout (CDNA5 ISA): A: lane l holds M=l%16, K=2*(l/16)+{0,1};
// B mirrored with N=l%16; D: VGPR r holds M=r+8*(l/16), N=l%16.
// ---------------------------------------------------------------------------
__global__ void __launch_bounds__(256)
gt_qkv_proj_wmma(const float* __restrict__ X,
                 const float* __restrict__ Wq,
                 const float* __restrict__ Wk,
                 const float* __restrict__ Wv,
                 float* __restrict__ Q,
                 float* __restrict__ K,
                 float* __restrict__ V)
{
    const int wave = (int)((blockIdx.x * blockDim.x + threadIdx.x) >> 5);
    const int lane = (int)(threadIdx.x & 31);
    const int NTASK = (N_NODES / 16) * 3;          // 9375 wave tasks
    if (wave >= NTASK) return;                     // uniform per wave -> scalar branch

    const int mat  = wave % 3;
    const int row0 = (wave / 3) * 16;
    const float* __restrict__ W   = (mat == 0) ? Wq : (mat == 1) ? Wk : Wv;
    float*       __restrict__ Out = (mat == 0) ? Q  : (mat == 1) ? K  : V;

    const int grp = lane >> 4;                     // 0: K pair {0,1}, 1: K pair {2,3}
    const int m16 = lane & 15;

    v8f c0 = {}, c1 = {}, c2 = {}, c3 = {};

    const float* __restrict__ arow = X + (size_t)(row0 + m16) * F_IN + 2 * grp;

#pragma unroll 4
    for (int k0 = 0; k0 < F_IN; k0 += 4) {
        v2f a;
        a.x = arow[k0];
        a.y = arow[k0 + 1];

        const float* __restrict__ wr0 = W + (size_t)(k0 + 2 * grp) * HD + m16;
        const float* __restrict__ wr1 = wr0 + HD;

        v2f b0, b1, b2, b3;
        b0.x = wr0[0];  b0.y = wr1[0];
        b1.x = wr0[16]; b1.y = wr1[16];
        b2.x = wr0[32]; b2.y = wr1[32];
        b3.x = wr0[48]; b3.y = wr1[48];

        c0 = __builtin_amdgcn_wmma_f32_16x16x4_f32(false, a, false, b0, (short)0, c0, false, false);
        c1 = __builtin_amdgcn_wmma_f32_16x16x4_f32(false, a, false, b1, (short)0, c1, false, false);
        c2 = __builtin_amdgcn_wmma_f32_16x16x4_f32(false, a, false, b2, (short)0, c2, false, false);
        c3 = __builtin_amdgcn_wmma_f32_16x16x4_f32(false, a, false, b3, (short)0, c3, false, false);
    }

    const int mrow = 8 * grp;
#pragma unroll
    for (int r = 0; r < 8; ++r) {
        float* __restrict__ o = Out + (size_t)(row0 + mrow + r) * HD + m16;
        o[0]  = c0[r];
        o[16] = c1[r];
        o[32] = c2[r];
        o[48] = c3[r];
    }
}

// ---------------------------------------------------------------------------
// Kernel 2: edge phase. One wave32 per edge iteration.
// Lane l owns flat (h,d) slots l and l+32 of the 64 per-edge values.
// Per-head dot product reduced across the 16 lanes of each head group
// via __shfl_xor; fp32 global atomics accumulate into d_out (wV) and z.
// All node arrays (~52 MB) are L2-resident on MI455X (192 MB L2).
// ---------------------------------------------------------------------------
__global__ void __launch_bounds__(256)
gt_edge_attn(const int* __restrict__ src, const int* __restrict__ dst,
             const float* __restrict__ Q, const float* __restrict__ K,
             const float* __restrict__ V,
             float* __restrict__ wV, float* __restrict__ z)
{
    const int lane  = (int)(threadIdx.x & 31);
    const int wave  = (int)((blockIdx.x * blockDim.x + threadIdx.x) >> 5);
    const int nwave = (int)((gridDim.x * blockDim.x) >> 5);
    const int grp   = lane >> 4;       // head group: heads {0,1} slot0, {2,3} slot1
    const int dd    = lane & 15;       // d within head

    for (int e = wave; e < E_EDGES; e += nwave) {
        const int s = __builtin_amdgcn_readfirstlane(src[e]);
        const int d = __builtin_amdgcn_readfirstlane(dst[e]);

        const size_t sb = (size_t)s * HD;
        const size_t db = (size_t)d * HD;

        const float k0 = K[sb + lane], k1 = K[sb + lane + 32];
        const float q0 = Q[db + lane], q1 = Q[db + lane + 32];
        const float v0 = V[sb + lane], v1 = V[sb + lane + 32];

        float p0 = k0 * q0;            // head = grp       (0 or 1)
        float p1 = k1 * q1;            // head = grp + 2   (2 or 3)

#pragma unroll
        for (int off = 1; off < 16; off <<= 1) {
            p0 += __shfl_xor(p0, off, 32);
            p1 += __shfl_xor(p1, off, 32);
        }

        // score / sqrt(D=16) = *0.25, clamp(+-5), exp
        const float s0 = __expf(fminf(fmaxf(p0 * 0.25f, -5.0f), 5.0f));
        const float s1 = __expf(fminf(fmaxf(p1 * 0.25f, -5.0f), 5.0f));

        atomicAdd(&wV[db + lane],      s0 * v0);
        atomicAdd(&wV[db + lane + 32], s1 * v1);
        if (dd == 0) {
            atomicAdd(&z[(size_t)d * NHEAD + grp],     s0);
            atomicAdd(&z[(size_t)d * NHEAD + 2 + grp], s1);
        }
    }
}

// ---------------------------------------------------------------------------
// Kernel 3: out = wV / (z + 1e-6), in place on d_out.
// ---------------------------------------------------------------------------
__global__ void __launch_bounds__(256)
gt_normalize(float* __restrict__ out, const float* __restrict__ z)
{
    const int i = (int)(blockIdx.x * blockDim.x + threadIdx.x);
    if (i < N_NODES * HD) {
        const int n = i >> 6;          // node
        const int h = (i >> 4) & 3;    // head
        out[i] = out[i] / (z[n * NHEAD + h] + 1e-6f);
    }
}

extern "C" void kernel_launch(void* const* d_in, const int* in_sizes, int n_in,
                              void* d_out, int out_size, void* d_ws, size_t ws_size,
                              hipStream_t stream)
{
    (void)in_sizes; (void)n_in; (void)out_size; (void)ws_size;

    const float* X   = (const float*)d_in[0];   // node_feats [N,128]
    const int*   src = (const int*)  d_in[1];   // [E]
    const int*   dst = (const int*)  d_in[2];   // [E]
    const float* Wq  = (const float*)d_in[3];   // [128,64]
    const float* Wk  = (const float*)d_in[4];
    const float* Wv  = (const float*)d_in[5];

    float* out = (float*)d_out;                 // [N,H,D] == [N,64]; doubles as wV accumulator
    float* ws  = (float*)d_ws;
    float* Q   = ws;                            // [N,64]
    float* K   = ws + (size_t)N_NODES * HD;     // [N,64]
    float* V   = ws + (size_t)2 * N_NODES * HD; // [N,64]
    float* z   = ws + (size_t)3 * N_NODES * HD; // [N,4]

    // zero accumulators (graph-capturable memset nodes)
    hipMemsetAsync(out, 0, (size_t)N_NODES * HD * sizeof(float), stream);
    hipMemsetAsync(z,   0, (size_t)N_NODES * NHEAD * sizeof(float), stream);

    // 1) projections (WMMA fp32)
    {
        const int ntask = (N_NODES / 16) * 3;   // 9375 waves
        const int wpb   = 256 / 32;             // 8 waves per block
        const int blocks = (ntask + wpb - 1) / wpb;
        gt_qkv_proj_wmma<<<blocks, 256, 0, stream>>>(X, Wq, Wk, Wv, Q, K, V);
    }

    // 2) edge attention + scatter
    gt_edge_attn<<<4096, 256, 0, stream>>>(src, dst, Q, K, V, out, z);

    // 3) normalize
    {
        const int total = N_NODES * HD;
        gt_normalize<<<(total + 255) / 256, 256, 0, stream>>>(out, z);
    }
}